// TPLCBlock_27341761806897
// MI455X (gfx1250) — compile-verified
//
#include <hip/hip_runtime.h>

// ---------------------------------------------------------------------------
// TimesNet-style block for MI455X (gfx1250, wave32, WMMA).
// Shapes fixed per reference: B=32, T=1024, C=512, top_k=3, dw kernel 3x3.
// Dominant compute: 3x pointwise GEMM (512x512 @ 512x32768) -> fp32 WMMA,
// now with double-buffered GLOBAL_LOAD_ASYNC_TO_LDS_B128 staging (ASYNCcnt).
// Workspace requirement: ~68 MB (header + FFT partials + ydw buffer).
// ---------------------------------------------------------------------------

typedef float v2f __attribute__((ext_vector_type(2)));
typedef float v8f __attribute__((ext_vector_type(8)));

#define B_    32
#define T_    1024
#define C_    512
#define BT_   (B_ * T_)    // 32768
#define NSEQ  (B_ * C_)    // 16384
#define NBIN  513          // rfft bins
#define NFFTWG 512         // FFT workgroups; each handles 32 sequences

// workspace layout in floats
#define WS_HDRI    0                       // 8 ints: periods p[3]
#define WS_HDRF    8                       // 8 floats: softmax weights w[3]
#define WS_PARTIAL 16                      // NFFTWG * NBIN partial amplitude sums
#define WS_YDW     (16 + NFFTWG * NBIN)    // BT_*C_ floats (16B aligned: 262672 % 4 == 0)

// ---------------------------------------------------------------------------
// CDNA5 async global->LDS copy (tracked with ASYNCcnt).
// VDST = per-lane LDS byte offset, VADDR = per-lane 64-bit global address.
// ---------------------------------------------------------------------------
__device__ __forceinline__ void async_ld_b128(unsigned ldsoff, const void* gptr)
{
    asm volatile("global_load_async_to_lds_b128 %0, %1, off"
                 :: "v"(ldsoff), "v"(gptr)
                 : "memory");
}

__device__ __forceinline__ void wait_async0()
{
    asm volatile("s_wait_asynccnt 0" ::: "memory");
}

// generic (flat) shared pointer -> wave-relative LDS byte offset (addr[31:0])
__device__ __forceinline__ unsigned lds_off(const void* p)
{
    return (unsigned)(unsigned long long)p;
}

// ---------------------------------------------------------------------------
// 1) out = x  (residual init)
// ---------------------------------------------------------------------------
__global__ __launch_bounds__(256)
void copy_x_kernel(const float4* __restrict__ x4, float4* __restrict__ out4, int n4)
{
    int i = blockIdx.x * 256 + threadIdx.x;
    if (i < n4) out4[i] = x4[i];
}

// ---------------------------------------------------------------------------
// 2) FFT amplitude: each WG does 32 sequences (same b, consecutive c so the
//    C-strided loads reuse L2 lines). Stockham radix-2, 1024-pt, in LDS.
//    Writes per-WG partial sums of |X[j]| (deterministic: no atomics).
// ---------------------------------------------------------------------------
__global__ __launch_bounds__(512)
void fft_amp_kernel(const float* __restrict__ x, float* __restrict__ partial)
{
    __shared__ float re0[1024], im0[1024], re1[1024], im1[1024];
    __shared__ float twr[1024], twi[1024];
    __shared__ float ampacc[NBIN];
    int tid = threadIdx.x;

    // twiddle table: index l+k holds exp(-i*pi*k/l), l = 2^stage
    for (int i = tid; i < 1024; i += 512) {
        if (i >= 1) {
            int l = 1 << (31 - __clz(i));
            int k = i - l;
            float ang = -3.14159265358979323846f * (float)k / (float)l;
            float s, c;
            sincosf(ang, &s, &c);
            twr[i] = c; twi[i] = s;
        } else { twr[i] = 1.f; twi[i] = 0.f; }
    }
    for (int j = tid; j < NBIN; j += 512) ampacc[j] = 0.f;
    __syncthreads();

    int b  = blockIdx.x >> 4;
    int c0 = (blockIdx.x & 15) << 5;

    for (int s = 0; s < 32; ++s) {
        int c = c0 + s;
        re0[tid]       = x[(size_t)(b * T_ + tid) * C_ + c];       im0[tid]       = 0.f;
        re0[tid + 512] = x[(size_t)(b * T_ + tid + 512) * C_ + c]; im0[tid + 512] = 0.f;
        __syncthreads();

        float *sr = re0, *si = im0, *dr = re1, *di = im1;
        for (int st = 0; st < 10; ++st) {
            int l  = 1 << st;
            int kk = tid & (l - 1);
            int j  = tid >> st;                 // tid == j*l + kk
            float wr = twr[l + kk], wi = twi[l + kk];
            float ar = sr[tid],        ai = si[tid];
            float br = sr[tid + 512],  bi = si[tid + 512];
            float tr = wr * br - wi * bi;
            float ti = wr * bi + wi * br;
            int o = 2 * j * l + kk;
            dr[o]     = ar + tr;  di[o]     = ai + ti;
            dr[o + l] = ar - tr;  di[o + l] = ai - ti;
            float* tp;
            tp = sr; sr = dr; dr = tp;
            tp = si; si = di; di = tp;
            __syncthreads();
        }
        float xr = sr[tid], xi = si[tid];
        ampacc[tid] += sqrtf(xr * xr + xi * xi);
        if (tid == 0) {
            float yr = sr[512], yi = si[512];
            ampacc[512] += sqrtf(yr * yr + yi * yi);
        }
        __syncthreads();
    }
    for (int j = tid; j < NBIN; j += 512)
        partial[(size_t)blockIdx.x * NBIN + j] = ampacc[j];
}

// ---------------------------------------------------------------------------
// 3) Fixed-order reduction + top-3 + softmax + periods -> header
// ---------------------------------------------------------------------------
__global__ __launch_bounds__(512)
void reduce_topk_kernel(const float* __restrict__ partial,
                        int* __restrict__ hdrI, float* __restrict__ hdrF)
{
    __shared__ float amp[NBIN];
    int tid = threadIdx.x;
    for (int j = tid; j < NBIN; j += 512) {
        float s = 0.f;
        for (int g = 0; g < NFFTWG; ++g) s += partial[(size_t)g * NBIN + j];
        amp[j] = s * (1.0f / (float)NSEQ);
    }
    __syncthreads();
    if (tid == 0) {
        amp[0] = 0.f;
        int bi[3]; float bv[3];
        for (int kk = 0; kk < 3; ++kk) {
            float best = -1.f; int besti = 1;
            for (int j = 0; j < NBIN; ++j) {
                bool skip = false;
                for (int m = 0; m < kk; ++m) if (bi[m] == j) skip = true;
                if (!skip && amp[j] > best) { best = amp[j]; besti = j; }
            }
            bi[kk] = besti; bv[kk] = best;
        }
        float mx = fmaxf(bv[0], fmaxf(bv[1], bv[2]));
        float e0 = expf(bv[0] - mx), e1 = expf(bv[1] - mx), e2 = expf(bv[2] - mx);
        float inv = 1.f / (e0 + e1 + e2);
        hdrF[0] = e0 * inv; hdrF[1] = e1 * inv; hdrF[2] = e2 * inv;
        for (int kk = 0; kk < 3; ++kk) {
            int idx = bi[kk];
            int p = (idx > 0) ? (T_ / idx) : T_;
            if (p < 1) p = 1;
            hdrI[kk] = p;
        }
    }
}

// ---------------------------------------------------------------------------
// 4) Depthwise 3x3 (SAME on n_rows x p grid, zeros beyond T). Output layout
//    ydw[bt][c] == [B,T,C]; reads & writes coalesced along c. p from header.
// ---------------------------------------------------------------------------
__global__ __launch_bounds__(256)
void dwconv_kernel(const float* __restrict__ x, const float* __restrict__ dw_w,
                   const float* __restrict__ dw_b, const int* __restrict__ hdrI,
                   float* __restrict__ ydw, int k)
{
    int idx = blockIdx.x * 256 + threadIdx.x;     // [0, BT_*C_)
    int c  = idx & (C_ - 1);
    int bt = idx >> 9;
    int t  = bt & (T_ - 1);
    int b  = bt >> 10;
    int p  = hdrI[k];
    int n_rows = (T_ + p - 1) / p;
    int r = t / p;
    int q = t - r * p;
    float acc = dw_b[c];
    #pragma unroll
    for (int dr = -1; dr <= 1; ++dr) {
        int rr = r + dr;
        if (rr < 0 || rr >= n_rows) continue;
        #pragma unroll
        for (int dq = -1; dq <= 1; ++dq) {
            int qq = q + dq;
            if (qq < 0 || qq >= p) continue;
            int tt = rr * p + qq;
            if (tt >= T_) continue;
            acc += dw_w[c * 9 + (dr + 1) * 3 + (dq + 1)]
                 * x[(size_t)(b * T_ + tt) * C_ + c];
        }
    }
    ydw[idx] = acc;
}

// ---------------------------------------------------------------------------
// 5) Pointwise GEMM via V_WMMA_F32_16X16X4_F32 (fp32 matrix path) with
//    double-buffered GLOBAL_LOAD_ASYNC_TO_LDS_B128 prefetch.
//    D[pos, cout] = sum_cin ydw[pos, cin] * pw_w[cout, cin]
//    out[pos*C + cout] += wk * (D + pw_b[cout])     (out preloaded with x)
//    WG tile: 64 pos x 256 cout; 8 waves as 2(pos) x 4(cout); each wave owns
//    32x64 = 2x4 grid of 16x16 accumulators. K staged in LDS in chunks of 16.
//    Row pad = 20 floats: 80B row stride keeps B128 LDS stores 16B-aligned
//    and makes all fragment reads bank-conflict-free (gcd(20,64) banks).
// ---------------------------------------------------------------------------
__global__ __launch_bounds__(256)
void pwgemm_kernel(const float* __restrict__ ydw, const float* __restrict__ pw_w,
                   const float* __restrict__ pw_b, const float* __restrict__ hdrF,
                   float* __restrict__ out, int k)
{
    __shared__ __attribute__((aligned(16))) float As[2][64][20];   // pos x k
    __shared__ __attribute__((aligned(16))) float Bs[2][256][20];  // cout x k (NOT transposed)

    int tid  = threadIdx.x;
    int lane = tid & 31, wave = tid >> 5;
    int wm = wave & 1, wn = wave >> 1;
    int pos0  = blockIdx.x * 64;
    int cout0 = blockIdx.y * 256;
    float wk = hdrF[k];

    // per-thread staging indices (constant across iterations)
    int api = tid >> 2, akq = (tid & 3) << 2;

    v8f acc[2][4];
    #pragma unroll
    for (int i = 0; i < 2; ++i)
        #pragma unroll
        for (int j = 0; j < 4; ++j)
            #pragma unroll
            for (int v = 0; v < 8; ++v) acc[i][j][v] = 0.f;

    // prologue: prefetch tile 0 into buffer 0
    {
        async_ld_b128(lds_off(&As[0][api][akq]),
                      ydw + (size_t)(pos0 + api) * C_ + 0 + akq);
        #pragma unroll
        for (int j = 0; j < 4; ++j) {
            int lin = tid + 256 * j;
            int cl = lin >> 2, kq = (lin & 3) << 2;
            async_ld_b128(lds_off(&Bs[0][cl][kq]),
                          pw_w + (size_t)(cout0 + cl) * C_ + 0 + kq);
        }
    }

    for (int it = 0; it < C_ / 16; ++it) {
        int cur = it & 1;
        wait_async0();       // my async loads into buf[cur] have landed
        __syncthreads();     // everyone's loads landed; prior reads of buf[cur^1] done

        if (it + 1 < C_ / 16) {
            int nk0 = (it + 1) * 16;
            async_ld_b128(lds_off(&As[1 - cur][api][akq]),
                          ydw + (size_t)(pos0 + api) * C_ + nk0 + akq);
            #pragma unroll
            for (int j = 0; j < 4; ++j) {
                int lin = tid + 256 * j;
                int cl = lin >> 2, kq = (lin & 3) << 2;
                async_ld_b128(lds_off(&Bs[1 - cur][cl][kq]),
                              pw_w + (size_t)(cout0 + cl) * C_ + nk0 + kq);
            }
        }

        int hi = lane >> 4, ln = lane & 15;
        #pragma unroll
        for (int kb = 0; kb < 16; kb += 4) {
            // A 16x4 layout: v0 = (M=ln, K=ka), v1 = (M=ln, K=ka+1); hi half K+=2
            int ka = kb + 2 * hi;
            v2f af[2], bf[4];
            #pragma unroll
            for (int mt = 0; mt < 2; ++mt) {
                int row = wm * 32 + mt * 16 + ln;
                af[mt][0] = As[cur][row][ka];
                af[mt][1] = As[cur][row][ka + 1];
            }
            #pragma unroll
            for (int nt = 0; nt < 4; ++nt) {
                int col = wn * 64 + nt * 16 + ln;
                bf[nt][0] = Bs[cur][col][ka];      // = pw_w[col][ka] = B[ka][col]
                bf[nt][1] = Bs[cur][col][ka + 1];
            }
            #pragma unroll
            for (int mt = 0; mt < 2; ++mt)
                #pragma unroll
                for (int nt = 0; nt < 4; ++nt)
                    acc[mt][nt] = __builtin_amdgcn_wmma_f32_16x16x4_f32(
                        false, af[mt], false, bf[nt],
                        (short)0, acc[mt][nt], false, false);
        }
        __syncthreads();     // done reading buf[cur]; safe to overwrite next round
    }

    // epilogue: C/D layout: VGPR v -> M = v + 8*hi, N = lane&15
    int hi = lane >> 4, ln = lane & 15;
    #pragma unroll
    for (int nt = 0; nt < 4; ++nt) {
        int cout = cout0 + wn * 64 + nt * 16 + ln;
        float bias = pw_b[cout];
        #pragma unroll
        for (int mt = 0; mt < 2; ++mt) {
            int rowb = pos0 + wm * 32 + mt * 16 + 8 * hi;
            #pragma unroll
            for (int v = 0; v < 8; ++v) {
                size_t a = (size_t)(rowb + v) * C_ + cout;
                out[a] += wk * (acc[mt][nt][v] + bias);
            }
        }
    }
}

// ---------------------------------------------------------------------------
extern "C" void kernel_launch(void* const* d_in, const int* in_sizes, int n_in,
                              void* d_out, int out_size, void* d_ws, size_t ws_size,
                              hipStream_t stream)
{
    const float* x    = (const float*)d_in[0];   // [B,T,C]
    const float* dw_w = (const float*)d_in[1];   // [C,1,3,3]
    const float* dw_b = (const float*)d_in[2];   // [C]
    const float* pw_w = (const float*)d_in[3];   // [C,C,1,1]
    const float* pw_b = (const float*)d_in[4];   // [C]
    float* out = (float*)d_out;

    float* ws      = (float*)d_ws;
    int*   hdrI    = (int*)ws;            // periods
    float* hdrF    = ws + WS_HDRF;        // softmax weights
    float* partial = ws + WS_PARTIAL;     // FFT partial sums
    float* ydw     = ws + WS_YDW;         // depthwise output [BT, C]

    int n4 = (BT_ * C_) / 4;
    copy_x_kernel<<<n4 / 256, 256, 0, stream>>>((const float4*)x, (float4*)out, n4);
    fft_amp_kernel<<<NFFTWG, 512, 0, stream>>>(x, partial);
    reduce_topk_kernel<<<1, 512, 0, stream>>>(partial, hdrI, hdrF);

    for (int k = 0; k < 3; ++k) {
        dwconv_kernel<<<(BT_ * C_) / 256, 256, 0, stream>>>(x, dw_w, dw_b, hdrI, ydw, k);
        dim3 g(BT_ / 64, C_ / 256);
        pwgemm_kernel<<<g, 256, 0, stream>>>(ydw, pw_w, pw_b, hdrF, out, k);
    }
}